// GraphNetwork_18751827214892
// MI455X (gfx1250) — compile-verified
//
#include <hip/hip_runtime.h>
#include <math.h>

typedef __attribute__((ext_vector_type(2))) float v2f;
typedef __attribute__((ext_vector_type(8))) float v8f;

#define NN 50000
#define NF 128
#define NH 128
#define NG 64
#define NC 10
#define LDA 132   // padded LDS row stride (floats): bank-conflict-free, keeps 16B align

// ---------------- degree kernels ----------------
__global__ void k_init_deg(float* __restrict__ deg, int n) {
    int i = blockIdx.x * blockDim.x + threadIdx.x;
    if (i < n) deg[i] = 1.0f;   // self-loop weight
}

__global__ void k_deg_scatter(const int* __restrict__ dst, float* __restrict__ deg, int E) {
    int e = blockIdx.x * blockDim.x + threadIdx.x;
    if (e < E) atomicAdd(&deg[dst[e]], 1.0f);
}

__global__ void k_deg_finalize(const float* __restrict__ deg, float* __restrict__ dis,
                               float* __restrict__ dinv, int n) {
    int i = blockIdx.x * blockDim.x + threadIdx.x;
    if (i < n) {
        float d = deg[i];
        dis[i]  = 1.0f / sqrtf(d);
        dinv[i] = 1.0f / d;
    }
}

// ---------------- WMMA fp32 GEMM: H[n,128] = act(X[n,128]) @ W[128,128] ----------------
// block = 256 threads = 8 waves; block covers 16 rows; wave w covers cols [w*16, w*16+16).
// A tile staged once in LDS (ELU fused), shared by all 8 waves.
template <bool APPLY_ELU>
__global__ void k_gemm128(const float* __restrict__ X, const float* __restrict__ W,
                          float* __restrict__ H, int nrows) {
    __shared__ float As[16 * LDA];
    const int row0 = blockIdx.x << 4;
    if (row0 >= nrows) return;

    // cooperative stage: 512 float4s, 2 per thread
    const float4* __restrict__ Xv = (const float4*)(X + (size_t)row0 * NF);
    for (int i = threadIdx.x; i < (16 * NF) / 4; i += 256) {
        float4 v = Xv[i];
        if (APPLY_ELU) {
            v.x = v.x > 0.0f ? v.x : expm1f(v.x);
            v.y = v.y > 0.0f ? v.y : expm1f(v.y);
            v.z = v.z > 0.0f ? v.z : expm1f(v.z);
            v.w = v.w > 0.0f ? v.w : expm1f(v.w);
        }
        int r = i >> 5;            // 32 float4 per row
        int c = (i & 31) << 2;
        *(float4*)&As[r * LDA + c] = v;   // 528B row stride -> stays 16B aligned
    }
    __syncthreads();

    const int lane = threadIdx.x & 31;
    const int wave = threadIdx.x >> 5;
    const int col0 = wave << 4;
    const int m  = lane & 15;           // M (for A) / N (for B,C)
    const int kh = (lane >> 4) << 1;    // lanes 0-15 -> K{0,1}, lanes 16-31 -> K{2,3}

    const float* __restrict__ arow = As + m * LDA + kh;

    v8f acc = {};
#pragma unroll
    for (int k0 = 0; k0 < NF; k0 += 4) {
        v2f A; A.x = arow[k0]; A.y = arow[k0 + 1];   // ds_load_b64, conflict-free
        v2f B;
        B.x = W[(size_t)(k0 + kh + 0) * NH + col0 + m];
        B.y = W[(size_t)(k0 + kh + 1) * NH + col0 + m];
        // V_WMMA_F32_16X16X4_F32 : D = A(16x4) x B(4x16) + C
        acc = __builtin_amdgcn_wmma_f32_16x16x4_f32(
            /*neg_a=*/false, A, /*neg_b=*/false, B,
            /*c_mod=*/(short)0, acc, /*reuse_a=*/false, /*reuse_b=*/false);
    }

    // C/D layout: VGPR v -> M = v + (lane<16 ? 0 : 8), N = lane&15
    const int mb = (lane >> 4) << 3;
#pragma unroll
    for (int v = 0; v < 8; ++v)
        H[(size_t)(row0 + mb + v) * NH + col0 + m] = acc[v];
}

// ---------------- AGG init: AGG = H * deg_inv + bias (self-loop + bias) ----------------
__global__ void k_agg_init(const float* __restrict__ H, const float* __restrict__ dinv,
                           const float* __restrict__ bias, float* __restrict__ AGG, int n) {
    int t = blockIdx.x * blockDim.x + threadIdx.x;   // one float4 per thread
    int node = t >> 5;
    if (node >= n) return;
    int fo = (t & 31) << 2;
    const float4 h = *(const float4*)(H + (size_t)node * NH + fo);
    const float4 b = *(const float4*)(bias + fo);
    float di = dinv[node];
    float4 r;
    r.x = h.x * di + b.x;
    r.y = h.y * di + b.y;
    r.z = h.z * di + b.z;
    r.w = h.w * di + b.w;
    *(float4*)(AGG + (size_t)node * NH + fo) = r;
}

// ---------------- edge scatter: AGG[dst] += H[src] * dis[src]*dis[dst] ----------------
// wave32 owns 32 consecutive edges: coalesced per-lane index loads, then __shfl
// broadcast; all 32 lanes move one float4 of payload per edge.
__global__ void k_edge_scatter(const int* __restrict__ src, const int* __restrict__ dst,
                               const float* __restrict__ dis, const float* __restrict__ H,
                               float* __restrict__ AGG, int E) {
    int lane = threadIdx.x & 31;
    int waveId = (blockIdx.x * blockDim.x + threadIdx.x) >> 5;
    int e0 = waveId << 5;
    if (e0 >= E) return;

    int s = 0, d = 0;
    float coef = 0.0f;
    int myE = e0 + lane;
    if (myE < E) {
        s = src[myE];
        d = dst[myE];
        coef = dis[s] * dis[d];
    }

    int fo = lane << 2;
    int nE = min(32, E - e0);
    for (int i = 0; i < nE; ++i) {
        int   si = __shfl(s, i, 32);
        int   di = __shfl(d, i, 32);
        float ci = __shfl(coef, i, 32);
        const float4 h = *(const float4*)(H + (size_t)si * NH + fo);
        float* __restrict__ o = AGG + (size_t)di * NH + fo;
        atomicAdd(o + 0, h.x * ci);
        atomicAdd(o + 1, h.y * ci);
        atomicAdd(o + 2, h.z * ci);
        atomicAdd(o + 3, h.w * ci);
    }
}

// ---------------- pooling ----------------
__global__ void k_pool_zero(float* __restrict__ sums, float* __restrict__ cnt) {
    int t = blockIdx.x * blockDim.x + threadIdx.x;
    if (t < NG * NH) sums[t] = 0.0f;
    if (t < NG) cnt[t] = 0.0f;
}

// wave32 owns 32 consecutive nodes, same broadcast pattern as edge scatter
__global__ void k_pool_scatter(const float* __restrict__ h, const int* __restrict__ batch,
                               float* __restrict__ sums, float* __restrict__ cnt, int n) {
    int lane = threadIdx.x & 31;
    int waveId = (blockIdx.x * blockDim.x + threadIdx.x) >> 5;
    int n0 = waveId << 5;
    if (n0 >= n) return;

    int g = 0;
    int myN = n0 + lane;
    if (myN < n) g = batch[myN];

    int fo = lane << 2;
    int nNodes = min(32, n - n0);
    for (int i = 0; i < nNodes; ++i) {
        int gi = __shfl(g, i, 32);
        const float4 v = *(const float4*)(h + (size_t)(n0 + i) * NH + fo);
        float* __restrict__ o = sums + (size_t)gi * NH + fo;
        atomicAdd(o + 0, v.x);
        atomicAdd(o + 1, v.y);
        atomicAdd(o + 2, v.z);
        atomicAdd(o + 3, v.w);
        if (lane == 0) atomicAdd(cnt + gi, 1.0f);
    }
}

__global__ void k_pool_finalize(const float* __restrict__ sums, const float* __restrict__ cnt,
                                float* __restrict__ out_pooled) {
    int t = blockIdx.x * blockDim.x + threadIdx.x;
    if (t >= NG * NH) return;
    int g = t >> 7;
    out_pooled[t] = sums[t] / fmaxf(cnt[g], 1.0f);
}

// ---------------- heads: cls = pooled@wc+bc, reg = pooled@wr+br ----------------
__global__ void k_heads(const float* __restrict__ pooled, const float* __restrict__ wc,
                        const float* __restrict__ bc, const float* __restrict__ wr,
                        const float* __restrict__ br, float* __restrict__ out_cls,
                        float* __restrict__ out_reg) {
    int t = blockIdx.x * blockDim.x + threadIdx.x;
    if (t < NG * NC) {
        int g = t / NC, c = t % NC;
        float s = bc[c];
        const float* p = pooled + (size_t)g * NH;
#pragma unroll 8
        for (int k = 0; k < NH; ++k) s += p[k] * wc[k * NC + c];
        out_cls[t] = s;
    } else if (t < NG * NC + NG) {
        int g = t - NG * NC;
        float s = br[0];
        const float* p = pooled + (size_t)g * NH;
#pragma unroll 8
        for (int k = 0; k < NH; ++k) s += p[k] * wr[k];
        out_reg[g] = s;
    }
}

extern "C" void kernel_launch(void* const* d_in, const int* in_sizes, int n_in,
                              void* d_out, int out_size, void* d_ws, size_t ws_size,
                              hipStream_t stream) {
    (void)n_in; (void)out_size; (void)ws_size;

    const float* x     = (const float*)d_in[0];
    const int*   ei    = (const int*)d_in[1];
    const int*   batch = (const int*)d_in[2];
    const float* w0 = (const float*)d_in[3];
    const float* b0 = (const float*)d_in[4];
    const float* w1 = (const float*)d_in[5];
    const float* b1 = (const float*)d_in[6];
    const float* w2 = (const float*)d_in[7];
    const float* b2 = (const float*)d_in[8];
    const float* wc = (const float*)d_in[9];
    const float* bc = (const float*)d_in[10];
    const float* wr = (const float*)d_in[11];
    const float* br = (const float*)d_in[12];

    const int E = in_sizes[1] / 2;
    const int* src = ei;
    const int* dst = ei + E;

    // workspace carve-out (all float, 16B-aligned chunks)
    float* deg  = (float*)d_ws;                       // NN
    float* dis  = deg + NN;                           // NN
    float* dinv = dis + NN;                           // NN
    float* sums = dinv + NN;                          // NG*NH
    float* cnt  = sums + NG * NH;                     // NG
    float* H    = cnt + NG;                           // NN*NH (offset stays 16B aligned)
    float* AGG  = H + (size_t)NN * NH;                // NN*NH

    float* out_pooled = (float*)d_out;                // NG*NH
    float* out_cls    = out_pooled + NG * NH;         // NG*NC
    float* out_reg    = out_cls + NG * NC;            // NG

    const int T = 256;
    const int gNodes    = (NN + T - 1) / T;
    const int gEdges    = (E + T - 1) / T;
    const int gNode32   = (NN * 32 + T - 1) / T;
    const int gGemm     = (NN + 15) / 16;
    const int gEdgeWave = (E + T - 1) / T;            // 8 waves/block, 32 edges/wave
    const int gPoolWave = (NN + T - 1) / T;           // 8 waves/block, 32 nodes/wave

    // degrees (self-loop included)
    k_init_deg<<<gNodes, T, 0, stream>>>(deg, NN);
    k_deg_scatter<<<gEdges, T, 0, stream>>>(dst, deg, E);
    k_deg_finalize<<<gNodes, T, 0, stream>>>(deg, dis, dinv, NN);

    // layer 0: H = x@w0 ; AGG = H*dinv + b0 ; edge scatter
    k_gemm128<false><<<gGemm, T, 0, stream>>>(x, w0, H, NN);
    k_agg_init<<<gNode32, T, 0, stream>>>(H, dinv, b0, AGG, NN);
    k_edge_scatter<<<gEdgeWave, T, 0, stream>>>(src, dst, dis, H, AGG, E);

    // layer 1: H = elu(AGG)@w1 ; AGG = H*dinv + b1 ; edge scatter
    k_gemm128<true><<<gGemm, T, 0, stream>>>(AGG, w1, H, NN);
    k_agg_init<<<gNode32, T, 0, stream>>>(H, dinv, b1, AGG, NN);
    k_edge_scatter<<<gEdgeWave, T, 0, stream>>>(src, dst, dis, H, AGG, E);

    // layer 2: H = elu(AGG)@w2 ; AGG = H*dinv + b2 ; edge scatter
    k_gemm128<true><<<gGemm, T, 0, stream>>>(AGG, w2, H, NN);
    k_agg_init<<<gNode32, T, 0, stream>>>(H, dinv, b2, AGG, NN);
    k_edge_scatter<<<gEdgeWave, T, 0, stream>>>(src, dst, dis, H, AGG, E);

    // global mean pool + heads
    k_pool_zero<<<(NG * NH + T - 1) / T, T, 0, stream>>>(sums, cnt);
    k_pool_scatter<<<gPoolWave, T, 0, stream>>>(AGG, batch, sums, cnt, NN);
    k_pool_finalize<<<(NG * NH + T - 1) / T, T, 0, stream>>>(sums, cnt, out_pooled);
    k_heads<<<(NG * NC + NG + T - 1) / T, T, 0, stream>>>(out_pooled, wc, bc, wr, br,
                                                          out_cls, out_reg);
}